// UnifiedEstimatorQNN_65481071398470
// MI455X (gfx1250) — compile-verified
//
#include <hip/hip_runtime.h>
#include <hip/hip_bf16.h>

typedef __attribute__((ext_vector_type(2))) float v2f;
typedef __attribute__((ext_vector_type(8))) float v8f;

#define B_ROWS 8192
#define F_DIM  128
#define Q_DIM  64
#define IN_DIM 193   // F + Q + 1
#define H_DIM  256
#define THRESH 0.95f

// workspace layout (float offsets)
#define WS_XN    0
#define WS_CNT   (WS_XN + B_ROWS * F_DIM)          // 8192*128
#define WS_FEATS (WS_CNT + B_ROWS)                 // +8192
#define WS_H1    (WS_FEATS + B_ROWS * IN_DIM)      // +8192*193
#define WS_H2    (WS_H1 + B_ROWS * H_DIM)          // +8192*256

// ---------------------------------------------------------------------------
// CDNA5 async global->LDS copy (8 bytes per lane), tracked by ASYNCcnt.
// Data moves memory->LDS without passing through VGPRs.
// ---------------------------------------------------------------------------
__device__ __forceinline__ void async_copy_b64(unsigned lds_off, const void* gaddr) {
    asm volatile("global_load_async_to_lds_b64 %0, %1, off"
                 :: "v"(lds_off), "v"(gaddr) : "memory");
}
__device__ __forceinline__ void wait_asynccnt0() {
    asm volatile("s_wait_asynccnt 0x0" ::: "memory");
}

// ---------------------------------------------------------------------------
// 1) Row L2-normalize x -> xn, and zero the adjacency counts.
// ---------------------------------------------------------------------------
__global__ void __launch_bounds__(256) norm_kernel(const float* __restrict__ x,
                                                   float* __restrict__ xn,
                                                   float* __restrict__ counts) {
    const int tid  = threadIdx.x;
    const int wave = tid >> 5;
    const int lane = tid & 31;
    const int row  = blockIdx.x * 8 + wave;

    const float4 v = *(const float4*)(x + (size_t)row * F_DIM + lane * 4);
    float s = v.x * v.x + v.y * v.y + v.z * v.z + v.w * v.w;
    #pragma unroll
    for (int off = 16; off >= 1; off >>= 1) s += __shfl_xor(s, off, 32);
    const float inv = 1.0f / (sqrtf(s) + 1e-12f);

    float4 o;
    o.x = v.x * inv; o.y = v.y * inv; o.z = v.z * inv; o.w = v.w * inv;
    *(float4*)(xn + (size_t)row * F_DIM + lane * 4) = o;

    const int gid = blockIdx.x * 256 + tid;
    if (gid < B_ROWS) counts[gid] = 0.0f;
}

// ---------------------------------------------------------------------------
// 2) Gram tile kernel: 128x128 tile of xn @ xn^T per workgroup, exact-f32 via
//    v_wmma_f32_16x16x4_f32.  LDS tiles filled with async global->LDS copies.
//    Epilogue thresholds fid=c^2 >= 0.95, drops diagonal, accumulates per-row
//    edge counts with one atomicAdd per row per tile (exact integer sums).
// ---------------------------------------------------------------------------
__global__ void __launch_bounds__(256) gram_kernel(const float* __restrict__ xn,
                                                   float* __restrict__ counts) {
    __shared__ float As[128][36];   // stride 36 floats: 8B-aligned, conflict-free
    __shared__ float Bs[128][36];

    const int tid   = threadIdx.x;
    const int wave  = tid >> 5;      // 0..7 -> 16-row block within tile
    const int lane  = tid & 31;
    const int lhalf = lane >> 4;     // selects K pair (A/B frag) / M+8 half (C)
    const int lmod  = lane & 15;     // M (A frag) / N (B,C frags)
    const int i0 = blockIdx.y * 128;
    const int j0 = blockIdx.x * 128;

    v8f acc[8] = {};                 // 8 col-tiles of 16: 16x16 f32 accumulators

    for (int k0 = 0; k0 < F_DIM; k0 += 32) {
        __syncthreads();             // previous chunk's compute done
        // async fill: 128 rows x 32 k per matrix, float2 per lane per issue
        #pragma unroll
        for (int it = 0; it < 8; ++it) {
            const int idx = tid + it * 256;     // 0..2047
            const int row = idx >> 4;
            const int v   = idx & 15;           // float2 index in 32-k chunk
            async_copy_b64((unsigned)(uintptr_t)&As[row][v * 2],
                           xn + (size_t)(i0 + row) * F_DIM + k0 + v * 2);
            async_copy_b64((unsigned)(uintptr_t)&Bs[row][v * 2],
                           xn + (size_t)(j0 + row) * F_DIM + k0 + v * 2);
        }
        wait_asynccnt0();            // own wave's async copies landed in LDS
        __syncthreads();             // all waves' copies visible

        #pragma unroll
        for (int kk = 0; kk < 32; kk += 4) {
            const v2f a = *(const v2f*)&As[16 * wave + lmod][kk + 2 * lhalf];
            #pragma unroll
            for (int ct = 0; ct < 8; ++ct) {
                const v2f b = *(const v2f*)&Bs[16 * ct + lmod][kk + 2 * lhalf];
                acc[ct] = __builtin_amdgcn_wmma_f32_16x16x4_f32(
                    false, a, false, b, (short)0, acc[ct], false, false);
            }
        }
    }

    // epilogue: fid = c^2, threshold, drop self-edges, reduce counts per row
    float cnt[8];
    #pragma unroll
    for (int r = 0; r < 8; ++r) cnt[r] = 0.0f;

    #pragma unroll
    for (int ct = 0; ct < 8; ++ct) {
        const int j = j0 + 16 * ct + lmod;
        #pragma unroll
        for (int r = 0; r < 8; ++r) {
            const int i = i0 + 16 * wave + r + 8 * lhalf;
            const float c = acc[ct][r];
            cnt[r] += ((c * c >= THRESH) && (i != j)) ? 1.0f : 0.0f;
        }
    }
    #pragma unroll
    for (int r = 0; r < 8; ++r) {
        float s = cnt[r];
        s += __shfl_xor(s, 8, 16);
        s += __shfl_xor(s, 4, 16);
        s += __shfl_xor(s, 2, 16);
        s += __shfl_xor(s, 1, 16);
        if (lmod == 0)
            atomicAdd(&counts[i0 + 16 * wave + r + 8 * lhalf], s);
    }
}

// ---------------------------------------------------------------------------
// 3) Build feats = [x | quantum | counts/8192]  (8192 x 193)
// ---------------------------------------------------------------------------
__global__ void feats_kernel(const float* __restrict__ x,
                             const float* __restrict__ q,
                             const float* __restrict__ counts,
                             float* __restrict__ feats) {
    const int gid = blockIdx.x * 256 + threadIdx.x;
    if (gid >= B_ROWS * IN_DIM) return;
    const int b = gid / IN_DIM;
    const int c = gid % IN_DIM;
    float v;
    if (c < F_DIM)              v = x[(size_t)b * F_DIM + c];
    else if (c < F_DIM + Q_DIM) v = q[(size_t)b * Q_DIM + (c - F_DIM)];
    else                        v = counts[b] * (1.0f / (float)B_ROWS);
    feats[gid] = v;
}

// ---------------------------------------------------------------------------
// 4) Generic WMMA-f32 GEMM: C[M,N] = act(A[M,K] @ W[N,K]^T + bias)
//    Tile 128(M) x 64(N), K chunked by 32 with zero-pad (handles K=193).
//    Global loads staged through register arrays so the compiler can clause
//    them under one s_wait_loadcnt, then burst the ds_stores.
// ---------------------------------------------------------------------------
__global__ void __launch_bounds__(256) gemm_bias_act(const float* __restrict__ A,
                                                     const float* __restrict__ W,
                                                     const float* __restrict__ bias,
                                                     float* __restrict__ C,
                                                     int M, int N, int K, int relu) {
    __shared__ float As[128][36];
    __shared__ float Ws[64][36];

    const int tid   = threadIdx.x;
    const int wave  = tid >> 5;
    const int lane  = tid & 31;
    const int lhalf = lane >> 4;
    const int lmod  = lane & 15;
    const int m0 = blockIdx.y * 128;
    const int n0 = blockIdx.x * 64;

    v8f acc[4] = {};

    for (int k0 = 0; k0 < K; k0 += 32) {
        // stage loads in registers (16 for A-tile, 8 for W-tile per thread)
        float a_reg[16];
        #pragma unroll
        for (int it = 0; it < 16; ++it) {
            const int idx = tid + it * 256;     // 128*32 = 4096 = 16*256
            const int row = idx >> 5, kk = idx & 31;
            const int k = k0 + kk;
            a_reg[it] = (k < K) ? A[(size_t)(m0 + row) * K + k] : 0.0f;
        }
        float w_reg[8];
        #pragma unroll
        for (int it = 0; it < 8; ++it) {
            const int idx = tid + it * 256;     // 64*32 = 2048 = 8*256
            const int row = idx >> 5, kk = idx & 31;
            const int k = k0 + kk;
            w_reg[it] = (k < K) ? W[(size_t)(n0 + row) * K + k] : 0.0f;
        }
        __syncthreads();             // previous chunk's compute done
        #pragma unroll
        for (int it = 0; it < 16; ++it) {
            const int idx = tid + it * 256;
            As[idx >> 5][idx & 31] = a_reg[it];
        }
        #pragma unroll
        for (int it = 0; it < 8; ++it) {
            const int idx = tid + it * 256;
            Ws[idx >> 5][idx & 31] = w_reg[it];
        }
        __syncthreads();

        #pragma unroll
        for (int kk = 0; kk < 32; kk += 4) {
            const v2f a = *(const v2f*)&As[16 * wave + lmod][kk + 2 * lhalf];
            #pragma unroll
            for (int ct = 0; ct < 4; ++ct) {
                const v2f b = *(const v2f*)&Ws[16 * ct + lmod][kk + 2 * lhalf];
                acc[ct] = __builtin_amdgcn_wmma_f32_16x16x4_f32(
                    false, a, false, b, (short)0, acc[ct], false, false);
            }
        }
    }

    #pragma unroll
    for (int ct = 0; ct < 4; ++ct) {
        const int gn = n0 + 16 * ct + lmod;
        const float bv = bias[gn];
        #pragma unroll
        for (int r = 0; r < 8; ++r) {
            const int gm = m0 + 16 * wave + r + 8 * lhalf;
            float c = acc[ct][r] + bv;
            if (relu) c = fmaxf(c, 0.0f);
            C[(size_t)gm * N + gn] = c;
        }
    }
}

// ---------------------------------------------------------------------------
// 5) Output layer: out[b] = h2[b,:] . W3 + b3   (wave per row)
// ---------------------------------------------------------------------------
__global__ void __launch_bounds__(256) out_kernel(const float* __restrict__ h2,
                                                  const float* __restrict__ W3,
                                                  const float* __restrict__ b3,
                                                  float* __restrict__ out) {
    const int tid  = threadIdx.x;
    const int wave = tid >> 5;
    const int lane = tid & 31;
    const int row  = blockIdx.x * 8 + wave;

    float s = 0.0f;
    #pragma unroll
    for (int i = 0; i < 8; ++i) {
        const int c = lane + i * 32;
        s += h2[(size_t)row * H_DIM + c] * W3[c];
    }
    #pragma unroll
    for (int off = 16; off >= 1; off >>= 1) s += __shfl_xor(s, off, 32);
    if (lane == 0) out[row] = s + b3[0];
}

// ---------------------------------------------------------------------------
extern "C" void kernel_launch(void* const* d_in, const int* in_sizes, int n_in,
                              void* d_out, int out_size, void* d_ws, size_t ws_size,
                              hipStream_t stream) {
    const float* x  = (const float*)d_in[0];
    const float* q  = (const float*)d_in[1];
    const float* W1 = (const float*)d_in[2];
    const float* b1 = (const float*)d_in[3];
    const float* W2 = (const float*)d_in[4];
    const float* b2 = (const float*)d_in[5];
    const float* W3 = (const float*)d_in[6];
    const float* b3 = (const float*)d_in[7];
    float* out = (float*)d_out;

    float* ws    = (float*)d_ws;
    float* xn    = ws + WS_XN;
    float* cnt   = ws + WS_CNT;
    float* feats = ws + WS_FEATS;
    float* h1    = ws + WS_H1;
    float* h2    = ws + WS_H2;

    // 1) normalize rows + zero counts
    norm_kernel<<<B_ROWS / 8, 256, 0, stream>>>(x, xn, cnt);
    // 2) Gram -> thresholded per-row edge counts (64x64 tiles of 128x128)
    gram_kernel<<<dim3(64, 64), 256, 0, stream>>>(xn, cnt);
    // 3) concat features
    feats_kernel<<<(B_ROWS * IN_DIM + 255) / 256, 256, 0, stream>>>(x, q, cnt, feats);
    // 4) MLP layers 1 & 2 (WMMA f32, fused bias+ReLU)
    gemm_bias_act<<<dim3(H_DIM / 64, B_ROWS / 128), 256, 0, stream>>>(
        feats, W1, b1, h1, B_ROWS, H_DIM, IN_DIM, 1);
    gemm_bias_act<<<dim3(H_DIM / 64, B_ROWS / 128), 256, 0, stream>>>(
        h1, W2, b2, h2, B_ROWS, H_DIM, H_DIM, 1);
    // 5) output projection
    out_kernel<<<B_ROWS / 8, 256, 0, stream>>>(h2, W3, b3, out);
}